// SAGPool_42674795053275
// MI455X (gfx1250) — compile-verified
//
#include <hip/hip_runtime.h>
#include <hip/hip_bf16.h>
#include <math.h>

#define N_NODES 100000
#define C_IN    256
#define N_EDGES 3200000
#define K_KEEP  80000

typedef __attribute__((ext_vector_type(2))) float v2f;
typedef __attribute__((ext_vector_type(8))) float v8f;

// ---------------------------------------------------------------------------
// Kernel 1: h = x @ W  via V_WMMA_F32_16X16X4_F32.
// One wave (32 lanes) computes 16 rows. B replicates W across all 16 columns,
// so every column of D holds the same matvec result; lanes 0 and 16 extract it.
// A 16x4 f32 layout: lanes 0-15 hold M=lane, VGPR0=K0 VGPR1=K1;
//                    lanes 16-31 hold M=lane-16, VGPR0=K2 VGPR1=K3.
// B 4x16 f32 layout mirrors A with N in place of M -> same index math.
// ---------------------------------------------------------------------------
__global__ void matvec_wmma_kernel(const float* __restrict__ x,
                                   const float* __restrict__ W,
                                   float* __restrict__ h) {
    const int wave = blockIdx.x * (blockDim.x >> 5) + (threadIdx.x >> 5);
    const int lane = threadIdx.x & 31;
    const int row0 = wave * 16;
    if (row0 >= N_NODES) return;              // wave-uniform exit: EXEC stays all-1s

    const int m    = lane & 15;               // row within tile (A) / column (B)
    const int koff = (lane >> 4) * 2;         // 0 for lanes 0-15, 2 for lanes 16-31
    const float* __restrict__ xrow = x + (size_t)(row0 + m) * C_IN;

    v8f acc = {};
    #pragma unroll 4
    for (int k0 = 0; k0 < C_IN; k0 += 4) {
        v2f a, b;
        a.x = xrow[k0 + koff];
        a.y = xrow[k0 + koff + 1];
        b.x = W[k0 + koff];
        b.y = W[k0 + koff + 1];
        acc = __builtin_amdgcn_wmma_f32_16x16x4_f32(
            /*neg_a=*/false, a, /*neg_b=*/false, b,
            /*c_mod=*/(short)0, acc, /*reuse_a=*/false, /*reuse_b=*/false);
    }

    // D layout: VGPR r, lanes 0-15 => M=r (N=lane); lanes 16-31 => M=8+r.
    // Lane 0 holds M=0..7 at N=0; lane 16 holds M=8..15 at N=0.
    if ((lane & 15) == 0) {
        const int mb = (lane >> 4) * 8;
        #pragma unroll
        for (int r = 0; r < 8; ++r) h[row0 + mb + r] = acc[r];
    }
}

// ---------------------------------------------------------------------------
// Degree / normalization / scoring
// ---------------------------------------------------------------------------
__global__ void zero_deg_kernel(int* __restrict__ degi) {
    int i = blockIdx.x * blockDim.x + threadIdx.x;
    if (i < N_NODES) degi[i] = 0;
}

__global__ void edge_deg_kernel(const int* __restrict__ ei, int* __restrict__ degi) {
    int e = blockIdx.x * blockDim.x + threadIdx.x;
    if (e < N_EDGES) atomicAdd(&degi[ei[N_EDGES + e]], 1);   // dst row
}

__global__ void node_prep_kernel(const float* __restrict__ h,
                                 const int* __restrict__ degi,
                                 const float* __restrict__ b,
                                 float* __restrict__ dinv,
                                 float* __restrict__ score) {
    int i = blockIdx.x * blockDim.x + threadIdx.x;
    if (i < N_NODES) {
        float dv = rsqrtf(1.0f + (float)degi[i]);
        dinv[i]  = dv;
        score[i] = dv * dv * h[i] + b[0];     // self-loop term + bias
    }
}

__global__ void edge_msg_kernel(const int* __restrict__ ei,
                                const float* __restrict__ dinv,
                                const float* __restrict__ h,
                                float* __restrict__ score) {
    int e = blockIdx.x * blockDim.x + threadIdx.x;
    if (e < N_EDGES) {
        int s = ei[e], d = ei[N_EDGES + e];
        atomicAdd(&score[d], dinv[s] * dinv[d] * h[s]);
    }
}

// ---------------------------------------------------------------------------
// Exact top-k rank by counting: rank[i] = #{j: s_j > s_i} + #{j<i: s_j == s_i}
// Matches lax.top_k ordering (descending value, ascending index on ties).
// ---------------------------------------------------------------------------
__global__ void rank_kernel(const float* __restrict__ score, int* __restrict__ rank) {
    __shared__ float tile[256];
    const int i  = blockIdx.x * 256 + threadIdx.x;
    const float si = (i < N_NODES) ? score[i] : 0.0f;
    int cnt = 0;
    for (int base = 0; base < N_NODES; base += 256) {
        int j = base + threadIdx.x;
        tile[threadIdx.x] = (j < N_NODES) ? score[j] : -INFINITY;
        __syncthreads();
        const int lim = min(256, N_NODES - base);
        if (lim == 256) {
            #pragma unroll 8
            for (int t = 0; t < 256; ++t) {
                float sj = tile[t];
                cnt += (int)((sj > si) | ((sj == si) & ((base + t) < i)));
            }
        } else {
            for (int t = 0; t < lim; ++t) {
                float sj = tile[t];
                cnt += (int)((sj > si) | ((sj == si) & ((base + t) < i)));
            }
        }
        __syncthreads();
    }
    if (i < N_NODES) rank[i] = cnt;
}

__global__ void build_sel_kernel(const float* __restrict__ score,
                                 const int* __restrict__ rank,
                                 int* __restrict__ idxmap,
                                 int* __restrict__ perm,
                                 float* __restrict__ topsc) {
    int i = blockIdx.x * blockDim.x + threadIdx.x;
    if (i < N_NODES) {
        int r = rank[i];
        if (r < K_KEEP) {
            idxmap[i] = r;
            perm[r]   = i;
            topsc[r]  = score[i];
        } else {
            idxmap[i] = -1;
        }
    }
}

// ---------------------------------------------------------------------------
// x_new = x[perm] * tanh(top_scores)   (one block of 64 lanes per kept row)
// ---------------------------------------------------------------------------
__global__ void gather_x_kernel(const float* __restrict__ x,
                                const int* __restrict__ perm,
                                const float* __restrict__ topsc,
                                float* __restrict__ o_x) {
    const int k = blockIdx.x;
    const int c = threadIdx.x;                 // 64 threads * float4 = 256 cols
    const int i = perm[k];
    const float g = tanhf(topsc[k]);
    const float4* __restrict__ src = (const float4*)(x + (size_t)i * C_IN);
    float4* __restrict__ dst = (float4*)(o_x + (size_t)k * C_IN);
    float4 v = src[c];
    v.x *= g; v.y *= g; v.z *= g; v.w *= g;
    dst[c] = v;
}

// ---------------------------------------------------------------------------
// Edge relabel + mask (ints emitted as exact float values)
// ---------------------------------------------------------------------------
__global__ void edge_out_kernel(const int* __restrict__ ei,
                                const int* __restrict__ idxmap,
                                float* __restrict__ o_edge,
                                float* __restrict__ o_mask) {
    int e = blockIdx.x * blockDim.x + threadIdx.x;
    if (e < N_EDGES) {
        int s  = ei[e], d = ei[N_EDGES + e];
        int ms = idxmap[s], md = idxmap[d];
        bool m = (ms >= 0) && (md >= 0);
        o_edge[e]           = m ? (float)ms : -1.0f;
        o_edge[N_EDGES + e] = m ? (float)md : -1.0f;
        o_mask[e]           = m ? 1.0f : 0.0f;
    }
}

__global__ void tail_out_kernel(const int* __restrict__ perm,
                                float* __restrict__ o_batch,
                                float* __restrict__ o_perm) {
    int k = blockIdx.x * blockDim.x + threadIdx.x;
    if (k < K_KEEP) {
        o_batch[k] = 0.0f;
        o_perm[k]  = (float)perm[k];
    }
}

// ---------------------------------------------------------------------------
extern "C" void kernel_launch(void* const* d_in, const int* in_sizes, int n_in,
                              void* d_out, int out_size, void* d_ws, size_t ws_size,
                              hipStream_t stream) {
    const float* x  = (const float*)d_in[0];   // [N, C]
    const int*   ei = (const int*)d_in[1];     // [2, E] row-major
    const float* W  = (const float*)d_in[2];   // [C]
    const float* b  = (const float*)d_in[3];   // [1]

    // Workspace partition (floats): 6*N + 2*K (rank/degi/idxmap/perm as ints) ~3.0 MB
    float* ws     = (float*)d_ws;
    float* h      = ws;                         // N
    int*   degi   = (int*)(ws + 1 * N_NODES);   // N
    float* dinv   = ws + 2 * N_NODES;           // N
    float* score  = ws + 3 * N_NODES;           // N
    int*   rank   = (int*)(ws + 4 * N_NODES);   // N
    int*   idxmap = (int*)(ws + 5 * N_NODES);   // N
    float* topsc  = ws + 6 * N_NODES;           // K
    int*   perm   = (int*)(ws + 6 * N_NODES + K_KEEP); // K

    // Output layout (all float32, outputs concatenated in return order)
    float* out     = (float*)d_out;
    float* o_x     = out;                                   // K*C
    float* o_edge  = o_x + (size_t)K_KEEP * C_IN;           // 2*E
    float* o_mask  = o_edge + 2 * (size_t)N_EDGES;          // E
    float* o_batch = o_mask + N_EDGES;                      // K
    float* o_perm  = o_batch + K_KEEP;                      // K

    const int nodeBlocks = (N_NODES + 255) / 256;
    const int edgeBlocks = (N_EDGES + 255) / 256;
    const int waveCount  = (N_NODES + 15) / 16;             // 6250 waves, 16 rows each
    const int mmBlocks   = (waveCount + 7) / 8;             // 8 waves per 256-thread block

    // 1) h = x @ W  (WMMA)
    matvec_wmma_kernel<<<mmBlocks, 256, 0, stream>>>(x, W, h);
    // 2) degrees (exact integer atomics)
    zero_deg_kernel<<<nodeBlocks, 256, 0, stream>>>(degi);
    edge_deg_kernel<<<edgeBlocks, 256, 0, stream>>>(ei, degi);
    // 3) dinv + self-loop/bias init of score
    node_prep_kernel<<<nodeBlocks, 256, 0, stream>>>(h, degi, b, dinv, score);
    // 4) neighbor message scatter-add
    edge_msg_kernel<<<edgeBlocks, 256, 0, stream>>>(ei, dinv, h, score);
    // 5) exact stable descending rank
    rank_kernel<<<nodeBlocks, 256, 0, stream>>>(score, rank);
    // 6) perm / top_scores / relabel map
    build_sel_kernel<<<nodeBlocks, 256, 0, stream>>>(score, rank, idxmap, perm, topsc);
    // 7) x_new = x[perm] * tanh(top_scores)
    gather_x_kernel<<<K_KEEP, 64, 0, stream>>>(x, perm, topsc, o_x);
    // 8) relabeled edges + edge mask
    edge_out_kernel<<<edgeBlocks, 256, 0, stream>>>(ei, idxmap, o_edge, o_mask);
    // 9) batch zeros + perm
    tail_out_kernel<<<(K_KEEP + 255) / 256, 256, 0, stream>>>(perm, o_batch, o_perm);
}